// QLSTM_65481071410734
// MI455X (gfx1250) — compile-verified
//
#include <hip/hip_runtime.h>

#define T_STEPS 256
#define BATCH   128
#define HIDDEN  512
#define NHEADS  8
#define HEADD   64

typedef __attribute__((ext_vector_type(2))) float v2f;
typedef __attribute__((ext_vector_type(8))) float v8f;

// workspace layout (float offsets)
#define WS_CNT  0        // unsigned barrier counter (zeroed by memset each launch)
#define WS_CF   4        // final cell-state scalar
#define WS_RQ   16       // rowsum(Wq)        [512]
#define WS_CTX  544      // ctx vector        [512]
#define WS_HW   1056     // current h vector  [512]
#define WS_H    2048     // h history         [256*512]
#define WS_K    133120   // K cache           [256*512]
#define WS_V    264192   // V cache           [256*512]

// agent-scope grid barrier for a fully-resident 8-block persistent kernel
__device__ __forceinline__ void gbarrier(unsigned* cnt, unsigned& epoch) {
  epoch += gridDim.x;
  __syncthreads();
  if (threadIdx.x == 0) {
    __hip_atomic_fetch_add(cnt, 1u, __ATOMIC_RELEASE, __HIP_MEMORY_SCOPE_AGENT);
    while (__hip_atomic_load(cnt, __ATOMIC_ACQUIRE, __HIP_MEMORY_SCOPE_AGENT) < epoch) {
      __builtin_amdgcn_s_sleep(1);
    }
  }
  __syncthreads();
}

// GEMV tile via f32 WMMA: out[16 rows] += W[rows, kb:ke] * x[kb:ke]
// A tile layout (ISA 16x4 f32): lanes 0-15 -> K=0,1 ; lanes 16-31 -> K=2,3.
// B is x broadcast across all 16 N-columns, so every D column equals the result.
__device__ __forceinline__ v8f gemv_tile(const float* __restrict__ wrow,
                                         const float* __restrict__ x,
                                         int kb, int ke, int kofs, v8f acc) {
#pragma unroll 4
  for (int k0 = kb; k0 < ke; k0 += 4) {
    v2f a = *(const v2f*)(wrow + k0 + kofs);
    v2f b = *(const v2f*)(x + k0 + kofs);
    acc = __builtin_amdgcn_wmma_f32_16x16x4_f32(false, a, false, b,
                                                (short)0, acc, false, false);
  }
  return acc;
}

__global__ void __launch_bounds__(512)
qlstm_recurrence(const float* __restrict__ ln_beta,
                 const float* __restrict__ attn_in_w,   // [1536,512] rows: Wq,Wk,Wv
                 const float* __restrict__ attn_in_b,   // [1536]
                 const float* __restrict__ attn_out_w,  // [512,512]
                 const float* __restrict__ attn_out_b,  // [512]
                 float* __restrict__ ws)
{
  __shared__ float q_lds[HEADD];
  __shared__ float sc[T_STEPS];        // scores / softmax weights for this head
  __shared__ float ctxp[8 * HEADD];    // ctx partial sums
  __shared__ float xbuf[HIDDEN];       // GEMV input vector
  __shared__ float p2[4 * 4 * 16];     // wmma partials phase2 (4 mtiles x 4 kslices)
  __shared__ float p3[8 * 2 * 16];     // wmma partials phase3 (8 mtiles x 2 kslices)

  unsigned* cnt  = (unsigned*)ws;
  float* rq      = ws + WS_RQ;
  float* ctx_ws  = ws + WS_CTX;
  float* h_ws    = ws + WS_HW;
  float* Hbuf    = ws + WS_H;
  float* Kc      = ws + WS_K;
  float* Vc      = ws + WS_V;

  const int tid  = threadIdx.x;
  const int blk  = blockIdx.x;          // == head index for phase 1
  const int lane = tid & 31;
  const int wave = tid >> 5;
  const int half = lane >> 4;
  const int kofs = half * 2;
  const int n64  = blk * HEADD;

  // LN over a singleton dim == beta exactly -> constant gate scalars
  const float fg = 1.f / (1.f + expf(-ln_beta[0]));
  const float ig = (1.f / (1.f + expf(-ln_beta[1]))) * tanhf(ln_beta[2]);
  const float og = 1.f / (1.f + expf(-ln_beta[3]));

  // one-time: rowsum(Wq) for this block's 64 rows (q_t = h~ * rq + bq)
  if (tid < 64) {
    const float* wr = attn_in_w + (blk * 64 + tid) * HIDDEN;
    float s = 0.f;
    for (int k = 0; k < HIDDEN; ++k) s += wr[k];
    rq[blk * 64 + tid] = s;
  }

  unsigned epoch = 0;
  float c = 0.f;

  for (int t = 0; t < T_STEPS; ++t) {
    c = fg * c + ig;                    // scalar LSTM cell
    const float ht = og * tanhf(c);     // hx_pre (same for every element)

    // ---- phase 1: this block owns head `blk`: scores, softmax, ctx ----
    if (t > 0) {
      if (tid < HEADD) q_lds[tid] = ht * rq[n64 + tid] + attn_in_b[n64 + tid];
      __syncthreads();
      for (int s = tid; s < t; s += 512) {
        const float* kr = Kc + s * HIDDEN + n64;
        float a = 0.f;
#pragma unroll 8
        for (int d = 0; d < HEADD; ++d) a += q_lds[d] * kr[d];
        sc[s] = a * 0.125f;             // 1/sqrt(64)
      }
      __syncthreads();
      if (tid < 32) {                   // wave 0: softmax over s < t
        float m = -3.0e38f;
        for (int s = tid; s < t; s += 32) m = fmaxf(m, sc[s]);
        for (int o = 16; o; o >>= 1) m = fmaxf(m, __shfl_xor(m, o, 32));
        float sum = 0.f;
        for (int s = tid; s < t; s += 32) { float e = expf(sc[s] - m); sc[s] = e; sum += e; }
        for (int o = 16; o; o >>= 1) sum += __shfl_xor(sum, o, 32);
        float inv = 1.f / sum;
        for (int s = tid; s < t; s += 32) sc[s] *= inv;
      }
      __syncthreads();
      {
        int cch = tid >> 6, d = tid & 63;
        float a = 0.f;
        for (int s = cch; s < t; s += 8) a += sc[s] * Vc[s * HIDDEN + n64 + d];
        ctxp[cch * HEADD + d] = a;
      }
      __syncthreads();
      if (tid < HEADD) {
        float a = 0.f;
#pragma unroll
        for (int cch = 0; cch < 8; ++cch) a += ctxp[cch * HEADD + tid];
        ctx_ws[n64 + tid] = a;
      }
    }
    gbarrier(cnt, epoch);               // full ctx visible to all blocks

    // ---- phase 2: h_t[blk*64 .. +64) = h~ + Wo*ctx + bo (WMMA GEMV) ----
    xbuf[tid] = (t > 0) ? ctx_ws[tid] : 0.f;
    __syncthreads();
    if (t > 0) {
      int mt = wave & 3, ks = wave >> 2;               // 4 mtiles x 4 k-slices(128)
      int row = blk * 64 + mt * 16 + (lane & 15);
      v8f acc = {0, 0, 0, 0, 0, 0, 0, 0};
      acc = gemv_tile(attn_out_w + row * HIDDEN, xbuf, ks * 128, ks * 128 + 128, kofs, acc);
      if ((lane & 15) == 0) {
#pragma unroll
        for (int i = 0; i < 8; ++i) p2[(mt * 4 + ks) * 16 + half * 8 + i] = acc[i];
      }
    }
    __syncthreads();
    if (tid < 64) {
      float hv = ht;
      if (t > 0) {
        int mt = tid >> 4, idx = tid & 15;
        float s = 0.f;
#pragma unroll
        for (int ks = 0; ks < 4; ++ks) s += p2[(mt * 4 + ks) * 16 + idx];
        hv += s + attn_out_b[blk * 64 + tid];
      }
      Hbuf[t * HIDDEN + blk * 64 + tid] = hv;
      h_ws[blk * 64 + tid] = hv;
    }
    gbarrier(cnt, epoch);               // full h_t visible

    // ---- phase 3: K_t/V_t = [Wk;Wv]*h_t + b (WMMA GEMV, 128 rows/block) ----
    xbuf[tid] = h_ws[tid];
    __syncthreads();
    {
      int mt = wave & 7, ks = wave >> 3;               // 8 mtiles x 2 k-slices(256)
      int grow = 512 + blk * 128 + mt * 16 + (lane & 15);
      v8f acc = {0, 0, 0, 0, 0, 0, 0, 0};
      acc = gemv_tile(attn_in_w + grow * HIDDEN, xbuf, ks * 256, ks * 256 + 256, kofs, acc);
      if ((lane & 15) == 0) {
#pragma unroll
        for (int i = 0; i < 8; ++i) p3[(mt * 2 + ks) * 16 + half * 8 + i] = acc[i];
      }
    }
    __syncthreads();
    if (tid < 128) {
      int mt = tid >> 4, idx = tid & 15;
      int rl = blk * 128 + tid;                        // 0..1023 across blocks
      float v = p3[(mt * 2 + 0) * 16 + idx] + p3[(mt * 2 + 1) * 16 + idx]
              + attn_in_b[512 + rl];
      if (rl < 512) Kc[t * HIDDEN + rl] = v;
      else          Vc[t * HIDDEN + (rl - 512)] = v;
    }
    gbarrier(cnt, epoch);               // K/V visible for next step
  }

  if (blk == 0 && tid == 0) ws[WS_CF] = c;
}

// outs[t,b,:] = h_t ; hx = h_{T-1} ; cx = c_final (all broadcast over B)
__global__ void __launch_bounds__(256)
qlstm_broadcast(const float4* __restrict__ H4, const float* __restrict__ cf,
                float4* __restrict__ out)
{
  const long long idx = (long long)blockIdx.x * 256 + threadIdx.x;
  const long long OUTS4 = (long long)T_STEPS * BATCH * (HIDDEN / 4);  // 4194304
  const long long HX4   = (long long)BATCH * (HIDDEN / 4);            // 16384
  if (idx < OUTS4) {
    int t  = (int)(idx >> 14);          // B*HID/4 = 16384 float4 per step
    int j4 = (int)(idx & 127);
    out[idx] = H4[t * (HIDDEN / 4) + j4];
  } else if (idx < OUTS4 + HX4) {
    int j4 = (int)((idx - OUTS4) & 127);
    out[idx] = H4[(T_STEPS - 1) * (HIDDEN / 4) + j4];
  } else if (idx < OUTS4 + 2 * HX4) {
    float cv = cf[0];
    out[idx] = make_float4(cv, cv, cv, cv);
  }
}

extern "C" void kernel_launch(void* const* d_in, const int* in_sizes, int n_in,
                              void* d_out, int out_size, void* d_ws, size_t ws_size,
                              hipStream_t stream) {
  (void)in_sizes; (void)n_in; (void)out_size; (void)ws_size;
  // d_in[0..4] (inputs, proj_w, proj_b, qparams, ln_gamma) are provably dead:
  // LayerNorm over a singleton dim returns beta exactly, zeroing the circuit.
  const float* ln_beta    = (const float*)d_in[5];
  const float* attn_in_w  = (const float*)d_in[6];
  const float* attn_in_b  = (const float*)d_in[7];
  const float* attn_out_w = (const float*)d_in[8];
  const float* attn_out_b = (const float*)d_in[9];
  float* ws = (float*)d_ws;

  hipMemsetAsync(d_ws, 0, 64, stream);  // reset barrier counter + c_final

  qlstm_recurrence<<<dim3(8), dim3(512), 0, stream>>>(
      ln_beta, attn_in_w, attn_in_b, attn_out_w, attn_out_b, ws);

  const int total4 = T_STEPS * BATCH * (HIDDEN / 4) + 2 * BATCH * (HIDDEN / 4);
  qlstm_broadcast<<<dim3((total4 + 255) / 256), dim3(256), 0, stream>>>(
      (const float4*)(ws + WS_H), ws + WS_CF, (float4*)d_out);
}